// LightweightAttention_87522843558382
// MI455X (gfx1250) — compile-verified
//
#include <hip/hip_runtime.h>
#include <hip/hip_bf16.h>

#define BB 32
#define SS 2048
#define DD 16
#define NEGV (-1e9f)

typedef __attribute__((ext_vector_type(2))) float v2f;
typedef __attribute__((ext_vector_type(4))) float v4f;
typedef __attribute__((ext_vector_type(8))) float v8f;

static __device__ __forceinline__ v8f wmma4(v2f a, v2f b, v8f c) {
  // D = A(16x4) x B(4x16) + C, fp32
  return __builtin_amdgcn_wmma_f32_16x16x4_f32(false, a, false, b, (short)0, c,
                                               false, false);
}

// Project a 16-row tile: out[r][e] = sum_d X[row0+r][d] * W[e][d] + bias[e]
// X: global row-major, stride 16 floats. Wl: LDS 16x16 row-major [e][d].
// out: LDS 16x16 row-major [r][e].
static __device__ __forceinline__ void proj_tile(const float* __restrict__ X,
                                                 long row0, const float* Wl,
                                                 const float* __restrict__ bias,
                                                 float* out, int lane) {
  const int l15  = lane & 15;            // M index for A, N (=e) index for B
  const int koff = (lane >> 4) << 1;     // K offset 0 or 2 (lane half)
  v8f acc = {};
#pragma unroll
  for (int t = 0; t < 4; ++t) {
    const int d0 = 4 * t + koff;
    const float* xr = X + (row0 + l15) * DD + d0;   // A(m=l15, k=d0..d0+1)
    v2f a; a.x = xr[0]; a.y = xr[1];
    const float* wr = Wl + l15 * DD + d0;           // B(k=d, n=e=l15) = W[e][d]
    v2f b; b.x = wr[0]; b.y = wr[1];
    acc = wmma4(a, b, acc);
  }
  const float bvl = bias[l15];
  const int mh = (lane >> 4) << 3;       // row half 0 or 8
#pragma unroll
  for (int r = 0; r < 8; ++r)
    out[(r + mh) * DD + l15] = acc[r] + bvl;        // D: (m=r+mh, n=l15)
}

__global__ __launch_bounds__(256) void attn_fused(
    const float* __restrict__ Q, const float* __restrict__ K,
    const float* __restrict__ V, const float* __restrict__ Wq,
    const float* __restrict__ Wk, const float* __restrict__ Wv,
    const float* __restrict__ bq, const float* __restrict__ bk,
    const float* __restrict__ bv, float* __restrict__ out,
    float* __restrict__ wout) {
  extern __shared__ float smem[];
  float* sWq  = smem;            // 256
  float* sWk  = smem + 256;      // 256
  float* sWv  = smem + 512;      // 256
  float* qt   = smem + 768;      // 256   (projected q tile)
  float* kv   = smem + 1024;     // 8*256 (per-wave k/v tile scratch)
  float* red  = smem + 3072;     // 256   (reduction scratch)
  float* rmax = smem + 3328;     // 16
  float* rsum = smem + 3344;     // 16
  float* ored = smem + 3360;     // 8*256 (output cross-wave reduce)
  float* s    = smem + 5408;     // 16*2048 score/weight strip

  const int qtile = blockIdx.x;
  const int b     = blockIdx.y;
  const int tid   = threadIdx.x;
  const int lane  = tid & 31;
  const int w     = tid >> 5;

  // Stage the three 16x16 weight matrices into LDS.
  for (int i = tid; i < 768; i += 256)
    smem[i] = (i < 256) ? Wq[i] : (i < 512 ? Wk[i - 256] : Wv[i - 512]);
  __syncthreads();

  const long base = (long)b * SS;
  if (w == 0) proj_tile(Q, base + (long)qtile * 16, sWq, bq, qt, lane);
  __syncthreads();

  const int l15  = lane & 15;
  const int koff = (lane >> 4) << 1;
  const int mh   = (lane >> 4) << 3;

  // ---- Phase 1: score tiles (wave w handles j = w, w+8, ...) ----
  for (int j = w; j <= qtile; j += 8) {
    float* kb = kv + w * 256;
    proj_tile(K, base + (long)j * 16, sWk, bk, kb, lane);
    v8f acc = {};
#pragma unroll
    for (int t = 0; t < 4; ++t) {
      const int d0 = 4 * t + koff;
      v2f a;  a.x  = qt[l15 * DD + d0]; a.y  = qt[l15 * DD + d0 + 1];
      v2f bf; bf.x = kb[l15 * DD + d0]; bf.y = kb[l15 * DD + d0 + 1];
      acc = wmma4(a, bf, acc);
    }
#pragma unroll
    for (int r = 0; r < 8; ++r) {
      const int mm = r + mh;
      const int qg = qtile * 16 + mm;
      const int kg = j * 16 + l15;
      float val = acc[r] * 0.25f;            // 1/sqrt(16)
      if (kg > qg) val = NEGV;               // causal mask
      s[mm * SS + kg] = val;
    }
  }
  __syncthreads();

  // ---- Phase 2: block softmax over 16 rows x ncols, b128 weight writes ----
  const int row   = tid >> 4;        // 16 threads per row
  const int c0    = tid & 15;        // chunk lane (float4 chunks)
  const int ncols = (qtile + 1) * 16;
  const int nch   = ncols >> 2;      // valid float4 chunks (multiple of 4)
  v4f* srow4 = (v4f*)(s + row * SS);

  float pm = -3.4e38f;
  for (int ch = c0; ch < nch; ch += 16) {
    const v4f v = srow4[ch];                         // ds_load_b128
    pm = fmaxf(pm, fmaxf(fmaxf(v.x, v.y), fmaxf(v.z, v.w)));
  }
  red[tid] = pm;
  __syncthreads();
  if (c0 == 0) {
    float mx0 = red[row * 16];
    for (int i = 1; i < 16; ++i) mx0 = fmaxf(mx0, red[row * 16 + i]);
    rmax[row] = mx0;
  }
  __syncthreads();
  const float mx = rmax[row];
  float ps = 0.f;
  for (int ch = c0; ch < nch; ch += 16) {
    v4f v = srow4[ch];
    v.x = __expf(v.x - mx);   // masked -> exp(-1e9-mx) underflows to 0
    v.y = __expf(v.y - mx);
    v.z = __expf(v.z - mx);
    v.w = __expf(v.w - mx);
    srow4[ch] = v;                                   // ds_store_b128
    ps += (v.x + v.y) + (v.z + v.w);
  }
  red[tid] = ps;
  __syncthreads();
  if (c0 == 0) {
    float sm = 0.f;
    for (int i = 0; i < 16; ++i) sm += red[row * 16 + i];
    rsum[row] = sm;
  }
  __syncthreads();
  const float inv = 1.0f / rsum[row];
  const int qg2 = qtile * 16 + row;
  v4f* wrow4 = (v4f*)(wout + ((long)b * SS + qg2) * SS);
  for (int ch = c0; ch < nch; ch += 16) {
    v4f v = srow4[ch] * inv;
    srow4[ch] = v;                                   // keep W in LDS for P3
    __builtin_nontemporal_store(v, wrow4 + ch);      // NT global_store_b128
  }
  const v4f z = {};
  for (int ch = nch + c0; ch < (SS >> 2); ch += 16)
    __builtin_nontemporal_store(z, wrow4 + ch);      // upper triangle = 0
  __syncthreads();

  // ---- Phase 3: output = W @ v (v projected on demand) ----
  v8f oacc = {};
  for (int j = w; j <= qtile; j += 8) {
    float* vb = kv + w * 256;
    proj_tile(V, base + (long)j * 16, sWv, bv, vb, lane);
#pragma unroll
    for (int t = 0; t < 4; ++t) {
      const int n0 = 4 * t + koff;
      v2f a;  a.x  = s[l15 * SS + j * 16 + n0];
              a.y  = s[l15 * SS + j * 16 + n0 + 1];
      v2f bf; bf.x = vb[n0 * DD + l15];
              bf.y = vb[(n0 + 1) * DD + l15];
      oacc = wmma4(a, bf, oacc);
    }
  }
#pragma unroll
  for (int r = 0; r < 8; ++r) ored[w * 256 + (r + mh) * 16 + l15] = oacc[r];
  __syncthreads();
  {
    float sum = 0.f;
#pragma unroll
    for (int ww = 0; ww < 8; ++ww) sum += ored[ww * 256 + tid];
    const int om = tid >> 4, od = tid & 15;
    out[((long)b * SS + qtile * 16 + om) * DD + od] = sum;
  }
}

extern "C" void kernel_launch(void* const* d_in, const int* in_sizes, int n_in,
                              void* d_out, int out_size, void* d_ws,
                              size_t ws_size, hipStream_t stream) {
  (void)in_sizes; (void)n_in; (void)d_ws; (void)ws_size; (void)out_size;
  const float* Q  = (const float*)d_in[0];
  const float* K  = (const float*)d_in[1];
  const float* V  = (const float*)d_in[2];
  const float* Wq = (const float*)d_in[3];
  const float* Wk = (const float*)d_in[4];
  const float* Wv = (const float*)d_in[5];
  const float* bq = (const float*)d_in[6];
  const float* bk = (const float*)d_in[7];
  const float* bv = (const float*)d_in[8];
  float* out  = (float*)d_out;                       // (B,S,16)
  float* wout = out + (long)BB * SS * DD;            // (B,S,S) follows

  const size_t shmem = (size_t)(5408 + 16 * SS) * sizeof(float);  // ~149 KB
  dim3 grid(SS / 16, BB);   // (128 q-tiles, 32 batches)
  attn_fused<<<grid, 256, shmem, stream>>>(Q, K, V, Wq, Wk, Wv, bq, bk, bv,
                                           out, wout);
}